// DAttentionX_55937654063650
// MI455X (gfx1250) — compile-verified
//
#include <hip/hip_runtime.h>

#define HEADS 8
#define NSEQ  4096
#define CDIM  128
#define HD    16
#define LN_EPS 1e-5f

typedef __attribute__((ext_vector_type(2))) float v2f;
typedef __attribute__((ext_vector_type(8))) float v8f;

__device__ __forceinline__ v8f wmma_f32_16x16x4(v2f a, v2f b, v8f c) {
    // D = A(16x4 f32) * B(4x16 f32) + C(16x16 f32)
    return __builtin_amdgcn_wmma_f32_16x16x4_f32(
        /*neg_a=*/false, a, /*neg_b=*/false, b,
        /*c_mod=*/(short)0, c, /*reuse_a=*/false, /*reuse_b=*/false);
}

// ---------------------------------------------------------------------------
// Kernel 1: LayerNorm + QK projection (WMMA f32) + value pre-scale.
// Block = 256 threads (8 waves), handles 16 rows of x.
// ---------------------------------------------------------------------------
__global__ __launch_bounds__(256)
void ln_qk_proj_kernel(const float* __restrict__ x,    // [N][C]
                       const float* __restrict__ Ain,  // [N][H]
                       const float* __restrict__ Wqk,  // [C][2C]
                       const float* __restrict__ Wv,   // [1]
                       const float* __restrict__ lng,  // [C]
                       const float* __restrict__ lnb,  // [C]
                       float* __restrict__ q_ws,       // [H][N][HD]
                       float* __restrict__ k_ws,       // [H][N][HD]
                       float* __restrict__ v_ws)       // [H][N]
{
    __shared__ float xn[16][132];   // +4 pad: conflict-free A-fragment reads

    const int tid  = threadIdx.x;
    const int lane = tid & 31;
    const int wave = tid >> 5;
    const int n0   = blockIdx.x * 16;

    // --- LayerNorm: wave w normalizes rows 2w, 2w+1 (32 lanes x 4 elems) ---
    for (int rr = 0; rr < 2; ++rr) {
        const int row = wave * 2 + rr;
        const float4 xv = *(const float4*)(x + (size_t)(n0 + row) * CDIM + lane * 4);
        float s  = xv.x + xv.y + xv.z + xv.w;
        float s2 = xv.x * xv.x + xv.y * xv.y + xv.z * xv.z + xv.w * xv.w;
        #pragma unroll
        for (int m = 1; m < 32; m <<= 1) {
            s  += __shfl_xor(s,  m, 32);
            s2 += __shfl_xor(s2, m, 32);
        }
        const float mu   = s * (1.0f / CDIM);
        const float var  = s2 * (1.0f / CDIM) - mu * mu;
        const float rstd = rsqrtf(var + LN_EPS);
        const int c = lane * 4;
        xn[row][c + 0] = (xv.x - mu) * rstd * lng[c + 0] + lnb[c + 0];
        xn[row][c + 1] = (xv.y - mu) * rstd * lng[c + 1] + lnb[c + 1];
        xn[row][c + 2] = (xv.z - mu) * rstd * lng[c + 2] + lnb[c + 2];
        xn[row][c + 3] = (xv.w - mu) * rstd * lng[c + 3] + lnb[c + 3];
    }
    __syncthreads();

    // --- Projection: each wave -> two 16x16 tiles of xn @ W_qk (K=128) ---
    const int col   = lane & 15;          // A row index == B col index
    const int khalf = (lane >> 4) * 2;    // K sub-offset per half-wave

    for (int tt = 0; tt < 2; ++tt) {
        const int tile = wave * 2 + tt;   // 0..15 -> cols [tile*16, tile*16+16)
        const int j0   = tile * 16;
        v8f acc = {};
        #pragma unroll 4
        for (int kk = 0; kk < 32; ++kk) {
            const int d0 = kk * 4 + khalf;
            const v2f a = *(const v2f*)&xn[col][d0];
            v2f b;
            b.x = Wqk[(size_t)d0       * (2 * CDIM) + j0 + col];
            b.y = Wqk[(size_t)(d0 + 1) * (2 * CDIM) + j0 + col];
            acc = wmma_f32_16x16x4(a, b, acc);
        }
        // qk.reshape(N,2,H,hd): j<128 -> q, else k; head = (j&127)>>4, d = j&15
        float* dst = (j0 < CDIM) ? q_ws : k_ws;
        const int h = (j0 & (CDIM - 1)) >> 4;
        const int rowbase = (lane >> 4) * 8;
        #pragma unroll
        for (int c = 0; c < 8; ++c) {
            const int row = rowbase + c;
            dst[((size_t)h * NSEQ + n0 + row) * HD + col] = acc[c];
        }
    }

    // --- v = A * W_v, stored head-major for coalesced reads in kernel 2 ---
    if (tid < 16 * HEADS) {
        const int r = tid >> 3, h = tid & 7;
        v_ws[(size_t)h * NSEQ + n0 + r] = Ain[(size_t)(n0 + r) * HEADS + h] * Wv[0];
    }
}

// ---------------------------------------------------------------------------
// Kernel 2: flash attention with rank-1 values, log2-domain online softmax
// with ONE v_exp_f32 per score element.
// One wave per (head, 16-query tile); 8 waves per block.
// ---------------------------------------------------------------------------
__global__ __launch_bounds__(256)
void attn_kernel(const float* __restrict__ q_ws,  // [H][N][HD]
                 const float* __restrict__ k_ws,  // [H][N][HD]
                 const float* __restrict__ v_ws,  // [H][N]
                 const float* __restrict__ Ain,   // [N][H]
                 float* __restrict__ out)         // [N][H]
{
    const int tid  = threadIdx.x;
    const int lane = tid & 31;
    const int gw   = blockIdx.x * 8 + (tid >> 5);
    const int h    = gw >> 8;          // 0..7
    const int qt   = gw & 255;         // 0..255
    const int n0   = qt * 16;
    // softmax in base-2: scale2 = C^-0.5 * log2(e); softmax invariant to base
    const float scale2 = 0.08838834764831845f * 1.4426950408889634f;

    const int col   = lane & 15;
    const int khalf = (lane >> 4) * 2;

    const float* qh = q_ws + (size_t)h * NSEQ * HD;
    const float* kh = k_ws + (size_t)h * NSEQ * HD;
    const float* vh = v_ws + (size_t)h * NSEQ;

    // Q fragments, pre-scaled so scores leave the WMMA already in log2 domain
    v2f a[4];
    #pragma unroll
    for (int kk = 0; kk < 4; ++kk) {
        v2f qa = *(const v2f*)(qh + (size_t)(n0 + col) * HD + kk * 4 + khalf);
        qa.x *= scale2; qa.y *= scale2;
        a[kk] = qa;
    }

    // Online-softmax state: each lane element owns one (row, col-subset)
    float runM[8], runD[8], runO[8];
    #pragma unroll
    for (int c = 0; c < 8; ++c) { runM[c] = -3.0e38f; runD[c] = 0.f; runO[c] = 0.f; }

    #pragma unroll 2
    for (int t = 0; t < NSEQ / 16; ++t) {
        const int m0 = t * 16;
        v8f acc = {};
        #pragma unroll
        for (int kk = 0; kk < 4; ++kk) {
            const v2f b = *(const v2f*)(kh + (size_t)(m0 + col) * HD + kk * 4 + khalf);
            acc = wmma_f32_16x16x4(a[kk], b, acc);   // S += (Q*scale2) * K^T
        }
        const float vv = vh[m0 + col];   // value for this lane's key column
        #pragma unroll
        for (int c = 0; c < 8; ++c) {
            const float s  = acc[c];
            const float d  = s - runM[c];
            // single transcendental: exp2(-|d|); -|d| folds into exp modifiers
            const float e  = __builtin_amdgcn_exp2f(-__builtin_fabsf(d));
            const bool  gt = d > 0.0f;
            const float mult = gt ? e : 1.0f;   // running side scaled if s is new max
            const float addD = gt ? 1.0f : e;
            const float addO = gt ? vv : e * vv;
            runD[c] = runD[c] * mult + addD;
            runO[c] = runO[c] * mult + addO;
            runM[c] = fmaxf(runM[c], s);
        }
    }

    // Merge 16 column-subsets per row (within each half-wave), same 1-exp trick
    #pragma unroll
    for (int m = 1; m < 16; m <<= 1) {
        #pragma unroll
        for (int c = 0; c < 8; ++c) {
            const float oM = __shfl_xor(runM[c], m, 32);
            const float oD = __shfl_xor(runD[c], m, 32);
            const float oO = __shfl_xor(runO[c], m, 32);
            const float d  = oM - runM[c];
            const float e  = __builtin_amdgcn_exp2f(-__builtin_fabsf(d));
            const bool  gt = d > 0.0f;
            const float mA = gt ? e : 1.0f;   // scales our side
            const float mB = gt ? 1.0f : e;   // scales the incoming side
            runD[c] = runD[c] * mA + oD * mB;
            runO[c] = runO[c] * mA + oO * mB;
            runM[c] = fmaxf(runM[c], oM);
        }
    }

    // C layout: VGPR c -> row c (lanes 0-15) / row c+8 (lanes 16-31)
    if (col == 0) {
        const int rb = (lane >> 4) * 8;
        #pragma unroll
        for (int c = 0; c < 8; ++c) {
            const int n = n0 + rb + c;
            out[(size_t)n * HEADS + h] = Ain[(size_t)n * HEADS + h] + runO[c] / runD[c];
        }
    }
}

// ---------------------------------------------------------------------------
extern "C" void kernel_launch(void* const* d_in, const int* in_sizes, int n_in,
                              void* d_out, int out_size, void* d_ws, size_t ws_size,
                              hipStream_t stream) {
    const float* x   = (const float*)d_in[0];   // [1,4096,128]
    const float* Ain = (const float*)d_in[1];   // [1,4096,8,1]
    const float* Wqk = (const float*)d_in[2];   // [128,256]
    const float* Wv  = (const float*)d_in[3];   // [1,1]
    const float* g   = (const float*)d_in[4];   // [128]
    const float* b   = (const float*)d_in[5];   // [128]
    float* out = (float*)d_out;                 // [1,4096,8,1]

    float* q_ws = (float*)d_ws;                       // H*N*HD floats
    float* k_ws = q_ws + (size_t)HEADS * NSEQ * HD;   // H*N*HD floats
    float* v_ws = k_ws + (size_t)HEADS * NSEQ * HD;   // H*N   floats

    ln_qk_proj_kernel<<<NSEQ / 16, 256, 0, stream>>>(x, Ain, Wqk, Wv, g, b,
                                                     q_ws, k_ws, v_ws);
    attn_kernel<<<(HEADS * (NSEQ / 16)) / 8, 256, 0, stream>>>(q_ws, k_ws, v_ws,
                                                               Ain, out);
}